// RetrievalModule_44341242364388
// MI455X (gfx1250) — compile-verified
//
#include <hip/hip_runtime.h>
#include <math.h>

// ---------------------------------------------------------------------------
// Retrieval: cosine-sim top-k + image gather for MI455X (gfx1250, wave32).
//   q:   [64, 512]  f32      ref: [50000, 512] f32
//   img: [50000, 3*32*32] f32  -> out [64, 5, 3*32*32] f32
// ---------------------------------------------------------------------------

typedef __attribute__((ext_vector_type(2))) float v2f;
typedef __attribute__((ext_vector_type(8))) float v8f;
typedef __attribute__((ext_vector_type(4))) int   v4i;

#define D        512
#define KTOP     5
#define IMG      3072          // 3*32*32
#define NT       64            // refs per block in sim kernel
#define DS_STRIDE 516          // 512 + 4 pad: conflict-free ds_load_b64 B-frags
#define NEG_BIG  (-1.0e30f)

// ---- gfx1250 async global->LDS copy (guarded; falls back to ld+st) --------
__device__ __forceinline__ void async_copy_b128(const float* gsrc, float* ldst) {
#if __has_builtin(__builtin_amdgcn_global_load_async_to_lds_b128)
  __builtin_amdgcn_global_load_async_to_lds_b128(
      (__attribute__((address_space(1))) v4i*)(gsrc),
      (__attribute__((address_space(3))) v4i*)(ldst),
      0, 0);
#else
  *(float4*)ldst = *(const float4*)gsrc;
#endif
}

__device__ __forceinline__ void wait_async0() {
#if __has_builtin(__builtin_amdgcn_s_wait_asynccnt)
  __builtin_amdgcn_s_wait_asynccnt(0);
#else
  asm volatile("s_wait_asynccnt 0" ::: "memory");
#endif
}

// ---------------------------------------------------------------------------
// Kernel 1: qinv[b] = 1 / max(||q_b||2, 1e-12)     (64 blocks x 1 wave)
// ---------------------------------------------------------------------------
__global__ void qnorm_kernel(const float* __restrict__ q, float* __restrict__ qinv) {
  const int b = blockIdx.x;
  const int lane = threadIdx.x;              // 0..31
  const float* r = q + (size_t)b * D;
  float s = 0.f;
#pragma unroll
  for (int i = 0; i < D / 32; ++i) { float v = r[lane + 32 * i]; s += v * v; }
  s += __shfl_xor(s, 16, 32);
  s += __shfl_xor(s, 8, 32);
  s += __shfl_xor(s, 4, 32);
  s += __shfl_xor(s, 2, 32);
  s += __shfl_xor(s, 1, 32);
  if (lane == 0) qinv[b] = 1.0f / fmaxf(sqrtf(s), 1e-12f);
}

// ---------------------------------------------------------------------------
// Kernel 2: sims[m][n] = (q_m . r_n) * qinv[m] * rinv[n] via V_WMMA_F32_16X16X4
// Block: 512 threads (16 waves) -> 64x64 output tile; grid.x = Npad/64.
// ---------------------------------------------------------------------------
__global__ void sim_kernel(const float* __restrict__ q,
                           const float* __restrict__ ref,
                           const float* __restrict__ qinv,
                           float* __restrict__ sims,
                           int N, int Npad) {
  extern __shared__ float smem[];
  float* sref = smem;                        // NT * DS_STRIDE floats
  float* rinv = smem + NT * DS_STRIDE;       // NT floats

  const int tid   = threadIdx.x;
  const int nbase = blockIdx.x * NT;

  // ---- async-load 64x512 f32 ref tile into LDS (16 b128 per thread) ----
#pragma unroll
  for (int j = 0; j < 16; ++j) {
    int i   = tid + 512 * j;                 // float4 index 0..8191
    int row = i >> 7;                        // /128 float4 per row
    int col = (i & 127) << 2;                // float column
    int g   = nbase + row; if (g > N - 1) g = N - 1;   // clamp (pad rows)
    async_copy_b128(ref + (size_t)g * D + col, sref + row * DS_STRIDE + col);
  }
  wait_async0();
  __syncthreads();

  // ---- ref row inverse norms from the LDS tile (8 lanes per row) ----
  {
    int row = tid >> 3;
    int sub = tid & 7;
    const float* rp = sref + row * DS_STRIDE + sub * 64;
    float s = 0.f;
#pragma unroll
    for (int i = 0; i < 64; ++i) s += rp[i] * rp[i];
    s += __shfl_xor(s, 1, 32);
    s += __shfl_xor(s, 2, 32);
    s += __shfl_xor(s, 4, 32);
    if (sub == 0) rinv[row] = 1.0f / fmaxf(sqrtf(s), 1e-12f);
  }
  __syncthreads();

  // ---- WMMA: wave -> one 16x16 tile of the 64x64 block ----
  const int wave   = tid >> 5;
  const int lane   = tid & 31;
  const int mt     = wave & 3;               // M tile (queries)
  const int nt     = wave >> 2;              // N tile (refs)
  const int laneHi = lane >> 4;              // 0: K=k0,k0+1   1: K=k0+2,k0+3
  const int l16    = lane & 15;
  const int kofs   = laneHi * 2;
  const int nloc   = nt * 16 + l16;

  const float* qrow = q    + (size_t)(mt * 16 + l16) * D + kofs;  // A frag base
  const float* brow = sref + nloc * DS_STRIDE + kofs;             // B frag base

  v8f c = {};
#pragma unroll 4
  for (int k0 = 0; k0 < D; k0 += 4) {
    float2 ta = *(const float2*)(qrow + k0);   // global_load_b64 (L2-resident)
    float2 tb = *(const float2*)(brow + k0);   // ds_load_b64, conflict-free
    v2f a = {ta.x, ta.y};
    v2f b = {tb.x, tb.y};
    c = __builtin_amdgcn_wmma_f32_16x16x4_f32(false, a, false, b,
                                              (short)0, c, false, false);
  }

  // ---- epilogue: fold both normalizations, pad rows -> -inf sentinel ----
  const float rs  = rinv[nloc];
  const int   ng  = nbase + nloc;
  const bool  oob = (ng >= N);
#pragma unroll
  for (int j = 0; j < 8; ++j) {
    int   m = mt * 16 + j + laneHi * 8;        // C/D VGPR layout
    float v = c[j] * qinv[m] * rs;
    if (oob) v = NEG_BIG;
    sims[(size_t)m * Npad + ng] = v;
  }
}

// ---------------------------------------------------------------------------
// Kernel 3: per-query top-5 + image gather. Grid = 64, block = 256.
// ---------------------------------------------------------------------------
__global__ void topk_gather_kernel(const float* __restrict__ sims,
                                   const float* __restrict__ images,
                                   float* __restrict__ out,
                                   int N, int Npad) {
  __shared__ float sv[256 * KTOP];
  __shared__ int   si[256 * KTOP];
  __shared__ int   fidx[KTOP];

  const int b   = blockIdx.x;
  const int tid = threadIdx.x;

  float bv[KTOP]; int bi[KTOP];
#pragma unroll
  for (int j = 0; j < KTOP; ++j) { bv[j] = NEG_BIG; bi[j] = 0; }

  const float* row = sims + (size_t)b * Npad;
  for (int n = tid; n < N; n += 256) {
    float v = row[n];
    if (v > bv[KTOP - 1]) {
      int p = KTOP - 1;
      while (p > 0 && v > bv[p - 1]) { bv[p] = bv[p - 1]; bi[p] = bi[p - 1]; --p; }
      bv[p] = v; bi[p] = n;
    }
  }
#pragma unroll
  for (int j = 0; j < KTOP; ++j) { sv[tid * KTOP + j] = bv[j]; si[tid * KTOP + j] = bi[j]; }
  __syncthreads();

  if (tid == 0) {
    float fv[KTOP]; int fi[KTOP];
#pragma unroll
    for (int j = 0; j < KTOP; ++j) { fv[j] = NEG_BIG; fi[j] = 0; }
    for (int t = 0; t < 256 * KTOP; ++t) {
      float v = sv[t];
      if (v > fv[KTOP - 1]) {
        int idx = si[t];
        int p = KTOP - 1;
        while (p > 0 && v > fv[p - 1]) { fv[p] = fv[p - 1]; fi[p] = fi[p - 1]; --p; }
        fv[p] = v; fi[p] = idx;
      }
    }
#pragma unroll
    for (int j = 0; j < KTOP; ++j) fidx[j] = fi[j];
  }
  __syncthreads();

  // gather KTOP images (float4-coalesced)
  for (int e = tid; e < KTOP * (IMG / 4); e += 256) {
    int j = e / (IMG / 4);
    int p = e % (IMG / 4);
    const float4* src = (const float4*)(images + (size_t)fidx[j] * IMG) + p;
    float4*       dst = (float4*)(out + ((size_t)b * KTOP + j) * IMG) + p;
    *dst = *src;
  }
}

// ---------------------------------------------------------------------------
extern "C" void kernel_launch(void* const* d_in, const int* in_sizes, int n_in,
                              void* d_out, int out_size, void* d_ws, size_t ws_size,
                              hipStream_t stream) {
  const float* q   = (const float*)d_in[0];
  const float* ref = (const float*)d_in[1];
  const float* img = (const float*)d_in[2];

  const int B    = in_sizes[0] / D;            // 64
  const int N    = in_sizes[1] / D;            // 50000
  const int Npad = ((N + NT - 1) / NT) * NT;   // 50048

  float* qinv = (float*)d_ws;                          // 64 floats
  float* sims = (float*)((char*)d_ws + 256);           // 64 * Npad floats

  qnorm_kernel<<<B, 32, 0, stream>>>(q, qinv);

  size_t smem = (size_t)(NT * DS_STRIDE + NT) * sizeof(float);   // ~129 KB LDS
  sim_kernel<<<Npad / NT, 512, smem, stream>>>(q, ref, qinv, sims, N, Npad);

  topk_gather_kernel<<<B, 256, 0, stream>>>(sims, img, (float*)d_out, N, Npad);
}